// BasePolicy_15659450761659
// MI455X (gfx1250) — compile-verified
//
#include <hip/hip_runtime.h>
#include <hip/hip_bf16.h>
#include <math.h>

// GATv2 forward for MI455X (gfx1250).
// GEMMs use V_WMMA_F32_16X16X4_F32 (exact fp32); edge phase is 3 streaming
// passes with float atomics (L2-resident gathers: xl+xr+out ~154MB < 192MB L2).

typedef __attribute__((ext_vector_type(2))) float v2f;
typedef __attribute__((ext_vector_type(8))) float v8f;

#define NNODES 50000
#define NEDGES 800000
#define FDIM   128
#define HC     256      // H*C
#define NHEAD  8
#define NEGSLOPE 0.2f

// ---- order-preserving float <-> uint encoding for atomic segment-max ----
__device__ __forceinline__ unsigned encf(float f) {
  unsigned u = __float_as_uint(f);
  return (u & 0x80000000u) ? ~u : (u | 0x80000000u);
}
__device__ __forceinline__ float decf(unsigned u) {
  return (u & 0x80000000u) ? __uint_as_float(u & 0x7fffffffu) : __uint_as_float(~u);
}

// ================= GEMM: xl = x@Wl, xr = x@Wr via fp32 WMMA =================
// One wave per 16x16 output tile. wave = rowTile*32 + q; q<16 -> Wl, else Wr.
// A (16x4 f32): lanes 0-15 M=lane,K=khalf..khalf+1 ; lanes 16-31 K=2..3
// B (4x16 f32): lanes 0-15 N=lane,K=khalf..        ; lanes 16-31 K=2..3
// D (16x16 f32): VGPR r -> M=r (lanes 0-15), M=r+8 (lanes 16-31); N=lane&15
__global__ __launch_bounds__(256) void gat_gemm_wmma(
    const float* __restrict__ x, const float* __restrict__ Wl,
    const float* __restrict__ Wr, float* __restrict__ xl, float* __restrict__ xr)
{
  const int wave = (blockIdx.x * blockDim.x + threadIdx.x) >> 5;
  const int lane = threadIdx.x & 31;
  const int rowTile = wave >> 5;           // 0..3124
  const int q       = wave & 31;           // 0..31
  const float* W = (q < 16) ? Wl : Wr;
  float*       Y = (q < 16) ? xl : xr;
  const int colBase = (q & 15) << 4;
  const int rowBase = rowTile << 4;

  const int mn    = lane & 15;             // M for A, N for B/D
  const int khalf = (lane >> 4) << 1;      // 0 or 2
  const float* xrow = x + (size_t)(rowBase + mn) * FDIM + khalf;
  const float* wcol = W + (size_t)khalf * HC + colBase + mn;

  v8f acc = {0.f, 0.f, 0.f, 0.f, 0.f, 0.f, 0.f, 0.f};
#pragma unroll 4
  for (int kk = 0; kk < FDIM; kk += 4) {
    v2f a, b;
    a.x = xrow[kk];
    a.y = xrow[kk + 1];
    b.x = wcol[(size_t)kk * HC];
    b.y = wcol[(size_t)(kk + 1) * HC];
    acc = __builtin_amdgcn_wmma_f32_16x16x4_f32(false, a, false, b,
                                                (short)0, acc, false, false);
  }
  const int mBase = rowBase + ((lane >> 4) << 3);
#pragma unroll
  for (int r = 0; r < 8; ++r)
    Y[(size_t)(mBase + r) * HC + colBase + mn] = acc[r];
}

// ============ self-loop attr: per-dst count & sum, then mean ============
__global__ void edge_degree(const int* __restrict__ dst,
                            const float* __restrict__ eattr,
                            float* __restrict__ cnt, float* __restrict__ asum)
{
  int e = blockIdx.x * blockDim.x + threadIdx.x;
  if (e >= NEDGES) return;
  int d = dst[e];
  atomicAdd(&cnt[d], 1.0f);
  atomicAdd(&asum[d], eattr[e]);
}

__global__ void node_mean(const float* __restrict__ cnt,
                          const float* __restrict__ asum,
                          float* __restrict__ loop_attr)
{
  int n = blockIdx.x * blockDim.x + threadIdx.x;
  if (n >= NNODES) return;
  loop_attr[n] = asum[n] / fmaxf(cnt[n], 1.0f);
}

__global__ void init_out(const float* __restrict__ bias, float* __restrict__ out)
{
  int t = blockIdx.x * blockDim.x + threadIdx.x;
  if (t >= NNODES * HC) return;
  out[t] = bias[t & (HC - 1)];
}

// ============ pass 1: per-edge logits + atomic segment max ============
// One wave per edge; lane handles 8 contiguous channels (one head per 4 lanes).
__global__ __launch_bounds__(256) void edge_logits(
    const int* __restrict__ src, const int* __restrict__ dst,
    const float* __restrict__ eattr, const float* __restrict__ loop_attr,
    const float* __restrict__ xl, const float* __restrict__ xr,
    const float* __restrict__ We, const float* __restrict__ att,
    float* __restrict__ logits, unsigned* __restrict__ maxkey)
{
  const int w = (blockIdx.x * blockDim.x + threadIdx.x) >> 5;
  const int lane = threadIdx.x & 31;
  if (w >= NEDGES + NNODES) return;   // uniform per wave
  int sid, did; float ea;
  if (w < NEDGES) { sid = src[w]; did = dst[w]; ea = eattr[w]; }
  else            { sid = w - NEDGES; did = sid; ea = loop_attr[sid]; }

  const int f0 = lane << 3;
  const int h  = lane >> 2;
  const float4* pl = (const float4*)(xl + (size_t)sid * HC + f0);
  const float4* pr = (const float4*)(xr + (size_t)did * HC + f0);
  const float4* pw = (const float4*)(We + f0);
  const float4* pa = (const float4*)(att + f0);

  float s = 0.f;
#pragma unroll
  for (int j = 0; j < 2; ++j) {
    float4 a = pl[j], b = pr[j], wv = pw[j], av = pa[j];
    float v;
    v = a.x + b.x + ea * wv.x; v = v > 0.f ? v : NEGSLOPE * v; s += av.x * v;
    v = a.y + b.y + ea * wv.y; v = v > 0.f ? v : NEGSLOPE * v; s += av.y * v;
    v = a.z + b.z + ea * wv.z; v = v > 0.f ? v : NEGSLOPE * v; s += av.z * v;
    v = a.w + b.w + ea * wv.w; v = v > 0.f ? v : NEGSLOPE * v; s += av.w * v;
  }
  s += __shfl_xor(s, 1, 32);   // reduce the 4 lanes of one head
  s += __shfl_xor(s, 2, 32);
  if ((lane & 3) == 0) {
    logits[(size_t)w * NHEAD + h] = s;
    atomicMax(&maxkey[did * NHEAD + h], encf(s));
  }
}

// ============ pass 2: ex = exp(logit - m); den += ex ============
__global__ void edge_exp(const int* __restrict__ dst,
                         float* __restrict__ logits,
                         const unsigned* __restrict__ maxkey,
                         float* __restrict__ den)
{
  int t = blockIdx.x * blockDim.x + threadIdx.x;
  if (t >= (NEDGES + NNODES) * NHEAD) return;
  int e = t >> 3, h = t & 7;
  int did = (e < NEDGES) ? dst[e] : (e - NEDGES);
  float m = decf(maxkey[did * NHEAD + h]);   // every segment has a self loop
  float ex = expf(logits[t] - m);
  logits[t] = ex;
  atomicAdd(&den[did * NHEAD + h], ex);
}

// ============ pass 3: out[dst] += (ex/den) * xl[src] ============
__global__ __launch_bounds__(256) void edge_scatter(
    const int* __restrict__ src, const int* __restrict__ dst,
    const float* __restrict__ exbuf, const float* __restrict__ den,
    const float* __restrict__ xl, float* __restrict__ out)
{
  const int w = (blockIdx.x * blockDim.x + threadIdx.x) >> 5;
  const int lane = threadIdx.x & 31;
  if (w >= NEDGES + NNODES) return;
  int sid, did;
  if (w < NEDGES) { sid = src[w]; did = dst[w]; }
  else            { sid = w - NEDGES; did = sid; }

  const int f0 = lane << 3;
  const int h  = lane >> 2;
  float alpha = exbuf[(size_t)w * NHEAD + h] / den[did * NHEAD + h];
  const float4* pl = (const float4*)(xl + (size_t)sid * HC + f0);
  float* po = out + (size_t)did * HC + f0;
  float4 a0 = pl[0], a1 = pl[1];
  atomicAdd(po + 0, alpha * a0.x);
  atomicAdd(po + 1, alpha * a0.y);
  atomicAdd(po + 2, alpha * a0.z);
  atomicAdd(po + 3, alpha * a0.w);
  atomicAdd(po + 4, alpha * a1.x);
  atomicAdd(po + 5, alpha * a1.y);
  atomicAdd(po + 6, alpha * a1.z);
  atomicAdd(po + 7, alpha * a1.w);
}

extern "C" void kernel_launch(void* const* d_in, const int* in_sizes, int n_in,
                              void* d_out, int out_size, void* d_ws, size_t ws_size,
                              hipStream_t stream) {
  const float* x     = (const float*)d_in[0];
  const int*   eidx  = (const int*)d_in[1];
  const int*   src   = eidx;
  const int*   dst   = eidx + NEDGES;
  const float* eattr = (const float*)d_in[2];
  const float* Wl    = (const float*)d_in[3];
  const float* Wr    = (const float*)d_in[4];
  const float* We    = (const float*)d_in[5];
  const float* att   = (const float*)d_in[6];
  const float* bias  = (const float*)d_in[7];
  float* out = (float*)d_out;

  // workspace layout (floats); total ~133.5 MB
  float* ws        = (float*)d_ws;
  float* xl        = ws;                                     // N*HC
  float* xr        = xl + (size_t)NNODES * HC;               // N*HC
  float* exbuf     = xr + (size_t)NNODES * HC;               // (E+N)*8
  float* loop_attr = exbuf + (size_t)(NEDGES + NNODES) * NHEAD;  // N
  float* cnt       = loop_attr + NNODES;                     // N
  float* asum      = cnt + NNODES;                           // N
  unsigned* maxkey = (unsigned*)(asum + NNODES);             // N*8
  float* den       = (float*)(maxkey + (size_t)NNODES * NHEAD); // N*8

  // zero only the accumulated state (cnt, asum, maxkey, den) — contiguous
  hipMemsetAsync(cnt, 0,
                 sizeof(float) * (size_t)NNODES * 2 +
                 sizeof(unsigned) * (size_t)NNODES * NHEAD +
                 sizeof(float) * (size_t)NNODES * NHEAD, stream);

  // GEMMs: (N/16) row tiles * 16 col tiles * 2 matrices = 100000 waves
  {
    const int waves = (NNODES / 16) * 32;
    gat_gemm_wmma<<<waves / 8, 256, 0, stream>>>(x, Wl, Wr, xl, xr);
  }
  edge_degree<<<(NEDGES + 255) / 256, 256, 0, stream>>>(dst, eattr, cnt, asum);
  node_mean<<<(NNODES + 255) / 256, 256, 0, stream>>>(cnt, asum, loop_attr);
  init_out<<<(NNODES * HC + 255) / 256, 256, 0, stream>>>(bias, out);

  const int TE = NEDGES + NNODES;             // 850000 edges incl. self loops
  const int waveBlocks = (TE + 7) / 8;        // 8 waves / 256-thread block
  edge_logits<<<waveBlocks, 256, 0, stream>>>(src, dst, eattr, loop_attr,
                                              xl, xr, We, att, exbuf, maxkey);
  edge_exp<<<(TE * NHEAD + 255) / 256, 256, 0, stream>>>(dst, exbuf, maxkey, den);
  edge_scatter<<<waveBlocks, 256, 0, stream>>>(src, dst, exbuf, den, xl, out);
}